// AdjacencyMatchingLoss_4879082849027
// MI455X (gfx1250) — compile-verified
//
#include <hip/hip_runtime.h>

// ---------------------------------------------------------------------------
// AdjacencyMatchingLoss, factored form:
//   S = (1/8) * sum_b  P[b] @ A @ P[b]^T        (A = (d_hw==1), 128x128)
//   out = - sum_e w[e]*S[i_e,j_e] / max(sum_e w[e], 1e-8)
// GEMMs use V_WMMA_F32_16X16X4_F32 (f32-in/f32-acc) to match f32 reference.
// A-matrix WMMA fragments are built ONCE per wave and reused across all 8
// batches (they live in ~64 VGPRs; wave32 allows up to 1024).
// ---------------------------------------------------------------------------

typedef __attribute__((ext_vector_type(2))) float v2f;
typedef __attribute__((ext_vector_type(8))) float v8f;

#define NLQ 128   // NL == NQ == 128
#define BB  8     // batch

// One block per 16-row band I of S.  8 waves (256 threads, wave32).
// Phase 1: Tband = P[b][band,:] @ A   (each wave -> one 16x16 tile of Tband)
// Phase 2: S_tile(I, wave) += Tband @ P[b]^T   (register accumulation over b)
__global__ __launch_bounds__(256)
void pap_gemm_kernel(const float* __restrict__ P,     // (8,128,128)
                     const int*   __restrict__ d_hw,  // (128,128)
                     float*       __restrict__ Smean) // (128,128) out
{
    __shared__ float Tband[16 * NLQ];    // 8 KB band of T = P[b] @ A

    const int lane = threadIdx.x & 31;
    const int wave = threadIdx.x >> 5;   // 0..7
    const int half = lane >> 4;          // K-pair selector (ISA 16x4 f32 layout)
    const int l16  = lane & 15;          // M (A-frag) / N (B-frag) index
    const int I    = blockIdx.x;         // band 0..7
    const int m0   = I * 16;
    const int n0   = wave * 16;          // this wave's tile-column base

    // ---- hoist: adjacency B-operand fragments for all 32 k-steps ----------
    // B(k, n) = (d_hw[k][n] == 1);  lane layout: n = l16, k-pair = 2*half.
    v2f afrag[32];
    {
        const int* dcol = d_hw + (n0 + l16);     // column (n0+l16), stride 128
        #pragma unroll
        for (int s = 0; s < 32; ++s) {
            const int k = 4 * s + 2 * half;
            afrag[s].x = (dcol[(k    ) * NLQ] == 1) ? 1.0f : 0.0f;
            afrag[s].y = (dcol[(k + 1) * NLQ] == 1) ? 1.0f : 0.0f;
        }
    }

    v8f sacc = {0.f, 0.f, 0.f, 0.f, 0.f, 0.f, 0.f, 0.f};

    for (int b = 0; b < BB; ++b) {
        const float* Pb = P + b * NLQ * NLQ;

        // ---------- phase 1: Tband = P[b][m0:m0+16, :] @ A ----------
        v8f tacc = {0.f, 0.f, 0.f, 0.f, 0.f, 0.f, 0.f, 0.f};
        {
            const float* arow = Pb + (m0 + l16) * NLQ;   // A-operand row (M=l16)
            #pragma unroll
            for (int s = 0; s < 32; ++s) {
                const int k = 4 * s + 2 * half;
                v2f af; af.x = arow[k]; af.y = arow[k + 1];
                tacc = __builtin_amdgcn_wmma_f32_16x16x4_f32(
                    false, af, false, afrag[s], (short)0, tacc, false, false);
            }
        }
        // C/D layout: VGPR r, lane -> (M = r + 8*half, N = l16)
        #pragma unroll
        for (int r = 0; r < 8; ++r)
            Tband[(r + 8 * half) * NLQ + (n0 + l16)] = tacc[r];
        __syncthreads();

        // ---------- phase 2: sacc += Tband @ P[b]^T  (tile rows m0, cols n0) --
        {
            const float* brow = Pb + (n0 + l16) * NLQ;   // B(k,n) = P[b][n0+n][k]
            #pragma unroll 8
            for (int s = 0; s < 32; ++s) {
                const int k = 4 * s + 2 * half;
                v2f af; af.x = Tband[l16 * NLQ + k]; af.y = Tband[l16 * NLQ + k + 1];
                v2f bf; bf.x = brow[k];              bf.y = brow[k + 1];
                sacc = __builtin_amdgcn_wmma_f32_16x16x4_f32(
                    false, af, false, bf, (short)0, sacc, false, false);
            }
        }
        __syncthreads();   // protect Tband before next b overwrites it
    }

    #pragma unroll
    for (int r = 0; r < 8; ++r)
        Smean[(m0 + r + 8 * half) * NLQ + (n0 + l16)] = sacc[r] * 0.125f;
}

// Single-block deterministic edge reduction (no float atomics).
__global__ __launch_bounds__(1024)
void edge_reduce_kernel(const int*   __restrict__ pairs,  // (E,2) int32
                        const float* __restrict__ w,      // (E,)
                        const float* __restrict__ Smean,  // (128,128)
                        float*       __restrict__ out,    // scalar
                        int E)
{
    __shared__ float sadj[1024];
    __shared__ float swsm[1024];

    float adj = 0.f, wsum = 0.f;
    for (int e = threadIdx.x; e < E; e += 1024) {
        const int i = pairs[2 * e + 0];
        const int j = pairs[2 * e + 1];
        const float we = w[e];
        adj  = fmaf(we, Smean[i * NLQ + j], adj);
        wsum += we;
    }
    sadj[threadIdx.x] = adj;
    swsm[threadIdx.x] = wsum;
    __syncthreads();
    #pragma unroll
    for (int off = 512; off > 0; off >>= 1) {
        if ((int)threadIdx.x < off) {
            sadj[threadIdx.x] += sadj[threadIdx.x + off];
            swsm[threadIdx.x] += swsm[threadIdx.x + off];
        }
        __syncthreads();
    }
    if (threadIdx.x == 0)
        out[0] = -sadj[0] / fmaxf(swsm[0], 1e-8f);
}

extern "C" void kernel_launch(void* const* d_in, const int* in_sizes, int n_in,
                              void* d_out, int out_size, void* d_ws, size_t ws_size,
                              hipStream_t stream)
{
    const float* P     = (const float*)d_in[0];   // (8,128,128) f32
    const int*   d_hw  = (const int*)  d_in[1];   // (128,128)  i32
    const int*   pairs = (const int*)  d_in[2];   // (E,2)      int
    const float* wts   = (const float*)d_in[3];   // (E,)       f32
    float*       out   = (float*)d_out;
    float*       Smean = (float*)d_ws;            // 128*128 f32 = 64 KB scratch
    const int    E     = in_sizes[3];

    pap_gemm_kernel<<<8, 256, 0, stream>>>(P, d_hw, Smean);
    edge_reduce_kernel<<<1, 1024, 0, stream>>>(pairs, wts, Smean, out, E);
}